// ResGCNBlock_61658550502017
// MI455X (gfx1250) — compile-verified
//
#include <hip/hip_runtime.h>
#include <hip/hip_bf16.h>

// RGCN basis-decomposition block for MI455X (gfx1250, wave32).
// Only the last layer's output survives in the reference -> compute layer L-1 only.
//
// Phase 1: xb = features @ V[L-1]  (fp32 WMMA 16x16x4, [M,64]x[64,256])
// Phase 2: per-edge msg = (comp[L-1][etype] . xb[src]) * norm, atomic scatter to h[dst]
// Phase 3: out = relu(h + bias[L-1]) + features

typedef __attribute__((ext_vector_type(2))) float v2f;
typedef __attribute__((ext_vector_type(8))) float v8f;

#define HD   64      // hidden dim
#define NB   4       // num bases
#define NCOL 256     // NB * HD  (xb row width)
#define LDSP 66      // padded LDS pitch (floats) to avoid bank conflicts

// ---------------------------------------------------------------------------
// Phase 1: xb[m][b*64+o] = sum_k feat[m][k] * V1[b][k][o]
// One wave -> one 16x16 C tile. Block = 8 waves covering 8 consecutive nTiles.
// grid.x = mTiles * 2 (nGroup 0 -> nTiles 0..7, nGroup 1 -> nTiles 8..15)
// ---------------------------------------------------------------------------
__global__ void __launch_bounds__(256) rgcn_proj_wmma(
    const float* __restrict__ A,    // [M,64] features
    const float* __restrict__ V1,   // [4,64,64] last-layer bases (b,k,o)
    float* __restrict__ XB,         // [M,256]
    int M)
{
    __shared__ float ldsA[16 * LDSP];

    const int mTile  = blockIdx.x >> 1;
    const int nGroup = blockIdx.x & 1;
    const int tid    = threadIdx.x;
    const int wave   = tid >> 5;
    const int lane   = tid & 31;

    // Stage the 16x64 A tile into LDS (each thread: one float4, rows guarded).
    {
        const int r  = tid >> 4;          // 0..15
        const int c4 = (tid & 15) * 4;    // 0,4,...,60
        const int m  = mTile * 16 + r;
        float4 v = make_float4(0.f, 0.f, 0.f, 0.f);
        if (m < M) v = *(const float4*)(A + (size_t)m * HD + c4);
        float* dp = &ldsA[r * LDSP + c4];
        dp[0] = v.x; dp[1] = v.y; dp[2] = v.z; dp[3] = v.w;
    }
    __syncthreads();

    const int nTile = nGroup * 8 + wave;          // 0..15
    const int bIdx  = nTile >> 2;                 // basis index 0..3
    const int mloc  = lane & 15;
    const int half  = lane >> 4;                  // 0: K pair {k,k+1}, 1: {k+2,k+3}
    const int oCol  = (nTile & 3) * 16 + mloc;    // output col within basis 0..63
    const float* Bb = V1 + (size_t)bIdx * (HD * HD) + oCol;  // stride HD in k

    v8f c = {};
#pragma unroll
    for (int k = 0; k < HD; k += 4) {
        const int k0 = k + half * 2;
        v2f a, b;
        // A fragment (ISA 32-bit A 16x4 layout): lane m=lane%16, VGPR0=K even pair
        a.x = ldsA[mloc * LDSP + k0];
        a.y = ldsA[mloc * LDSP + k0 + 1];
        // B fragment (4x16, row striped across lanes): n=lane%16, half selects K pair
        b.x = Bb[(size_t)(k0)     * HD];
        b.y = Bb[(size_t)(k0 + 1) * HD];
        c = __builtin_amdgcn_wmma_f32_16x16x4_f32(
                /*neg_a=*/false, a, /*neg_b=*/false, b,
                /*c_mod=*/(short)0, c, /*reuse_a=*/false, /*reuse_b=*/false);
    }

    // C/D layout: VGPR r -> row (r + 8*half), column = nTile*16 + lane%16
    const int nCol = nTile * 16 + mloc;
#pragma unroll
    for (int r = 0; r < 8; ++r) {
        const int m = mTile * 16 + r + half * 8;
        if (m < M) XB[(size_t)m * NCOL + nCol] = c[r];
    }
}

// ---------------------------------------------------------------------------
// Zero-init the accumulator h.
// ---------------------------------------------------------------------------
__global__ void zero_f4(float4* __restrict__ p, size_t n4)
{
    size_t i = (size_t)blockIdx.x * blockDim.x + threadIdx.x;
    if (i < n4) p[i] = make_float4(0.f, 0.f, 0.f, 0.f);
}

// ---------------------------------------------------------------------------
// Phase 2: one wave32 per edge; lane owns dims {2*lane, 2*lane+1}.
// msg = sum_b comp1[etype][b] * xb[src][b][:] ; h[dst] += msg * norm
// ---------------------------------------------------------------------------
__global__ void __launch_bounds__(256) edge_scatter(
    const float* __restrict__ xb,     // [N,256]
    const float* __restrict__ comp1,  // [32,4]
    const float* __restrict__ norm,   // [E]
    const int*   __restrict__ src,
    const int*   __restrict__ dst,
    const int*   __restrict__ etype,
    float* __restrict__ h,            // [N,64]
    int E)
{
    const int e    = (int)(((size_t)blockIdx.x * blockDim.x + threadIdx.x) >> 5);
    const int lane = threadIdx.x & 31;
    if (e >= E) return;   // wave-uniform exit

    const int   s = src[e];
    const int   d = dst[e];
    const int   r = etype[e];
    const float w = norm[e];
    const float4 cf = *(const float4*)(comp1 + r * NB);

    const float2* xs = (const float2*)(xb + (size_t)s * NCOL);
    const float2 v0 = xs[ 0 + lane];   // basis 0
    const float2 v1 = xs[32 + lane];   // basis 1
    const float2 v2 = xs[64 + lane];   // basis 2
    const float2 v3 = xs[96 + lane];   // basis 3

    float mx = cf.x * v0.x + cf.y * v1.x + cf.z * v2.x + cf.w * v3.x;
    float my = cf.x * v0.y + cf.y * v1.y + cf.z * v2.y + cf.w * v3.y;
    mx *= w; my *= w;

    float* hp = h + (size_t)d * HD + lane * 2;
    atomicAdd(hp,     mx);
    atomicAdd(hp + 1, my);
}

// ---------------------------------------------------------------------------
// Phase 3: out = relu(h + bias1) + features
// ---------------------------------------------------------------------------
__global__ void finalize(
    const float* __restrict__ h,
    const float* __restrict__ feat,
    const float* __restrict__ bias1,  // [64]
    float* __restrict__ out,
    size_t n)
{
    size_t i = (size_t)blockIdx.x * blockDim.x + threadIdx.x;
    if (i < n) {
        float v = h[i] + bias1[i & (HD - 1)];
        v = v > 0.f ? v : 0.f;
        out[i] = v + feat[i];
    }
}

// ---------------------------------------------------------------------------
extern "C" void kernel_launch(void* const* d_in, const int* in_sizes, int n_in,
                              void* d_out, int out_size, void* d_ws, size_t ws_size,
                              hipStream_t stream)
{
    const float* feat = (const float*)d_in[0];
    const float* norm = (const float*)d_in[1];
    const float* V    = (const float*)d_in[2];
    const float* comp = (const float*)d_in[3];
    const float* bias = (const float*)d_in[4];
    const int*   src  = (const int*)d_in[5];
    const int*   dst  = (const int*)d_in[6];
    const int*   et   = (const int*)d_in[7];
    float* out = (float*)d_out;

    const int M = in_sizes[0] / HD;                 // 100000 nodes
    const int E = in_sizes[5];                      // 1000000 edges
    const int L = in_sizes[2] / (NB * HD * HD);     // layers (2)

    const float* V1    = V    + (size_t)(L - 1) * NB * HD * HD;
    const float* comp1 = comp + (size_t)(L - 1) * (in_sizes[3] / L);
    const float* bias1 = bias + (size_t)(L - 1) * HD;

    float* xb = (float*)d_ws;                       // [M,256]  ~102.4 MB
    float* h  = xb + (size_t)M * NCOL;              // [M,64]   ~25.6 MB

    // Phase 1: WMMA projection
    const int mTiles = (M + 15) / 16;
    rgcn_proj_wmma<<<mTiles * 2, 256, 0, stream>>>(feat, V1, xb, M);

    // Zero accumulator
    const size_t nh = (size_t)M * HD;
    zero_f4<<<(unsigned)((nh / 4 + 255) / 256), 256, 0, stream>>>((float4*)h, nh / 4);

    // Phase 2: gather + atomic scatter (8 edges = 8 waves per block)
    edge_scatter<<<(E + 7) / 8, 256, 0, stream>>>(xb, comp1, norm, src, dst, et, h, E);

    // Phase 3: epilogue
    finalize<<<(unsigned)((nh + 255) / 256), 256, 0, stream>>>(h, feat, bias1, out, nh);
}